// ShapeCaps_27169963115018
// MI455X (gfx1250) — compile-verified
//
#include <hip/hip_runtime.h>
#include <math.h>

// Problem constants (from reference)
#define B_    32
#define O_    64
#define I_    1024
#define DIN   16
#define DOUT  32
#define M_    (O_ * DOUT)          // 2048 rows (o*32+d)
#define N_    B_                   // 32 cols (batch)
#define EPS_  1e-8f

// LDS u_hat tile: 2048 x 32 with padded row stride 33 to avoid bank conflicts
#define LDS_STRIDE 33
#define LDS_BYTES  (M_ * LDS_STRIDE * 4)   // 270,336 B < 320 KB WGP LDS

typedef __attribute__((ext_vector_type(2))) float v2f;
typedef __attribute__((ext_vector_type(8))) float v8f;

__global__ void caps_zero_kernel(float* __restrict__ p, int n) {
    int i = blockIdx.x * blockDim.x + threadIdx.x;
    if (i < n) p[i] = 0.0f;
}

// One block per input capsule i. 256 threads = 8 waves (wave32).
// Step 1: u_hat_i = W_i (2048x16) @ x_i^T (16x32) via v_wmma_f32_16x16x4_f32 -> LDS.
// Step 2: softmax over o (per (b,d)) using c[b,o,d]; accumulate s[b,o,d] over i via atomics.
__global__ __launch_bounds__(256)
void caps_route_pass_kernel(const float* __restrict__ x,   // [B, I, DIN]
                            const float* __restrict__ W,   // [O, I, DOUT, DIN]
                            const float* __restrict__ c,   // [B, O, DOUT]
                            float* __restrict__ s,         // [B, O, DOUT] (accumulated)
                            int iter)
{
    extern __shared__ float uhat[];   // [2048][LDS_STRIDE], col = b, row = o*32+d

    const int i       = blockIdx.x;
    const int tid     = threadIdx.x;
    const int lane    = tid & 31;
    const int wave    = tid >> 5;
    const int halfsel = lane >> 4;    // 0: lanes 0-15, 1: lanes 16-31
    const int l16     = lane & 15;

    // ---- B fragments from x: B[k, n] with n = batch. f32 4x16 per k-step:
    // VGPR v, lane L -> B[K = kk*4 + 2*halfsel + v, N = L&15]  => float2 load (consecutive k)
    v2f bf[2][4];
#pragma unroll
    for (int nt = 0; nt < 2; ++nt) {
        const int bidx = nt * 16 + l16;
        const float* xp = x + (size_t)bidx * (I_ * DIN) + (size_t)i * DIN + halfsel * 2;
#pragma unroll
        for (int kk = 0; kk < 4; ++kk)
            bf[nt][kk] = *(const v2f*)(xp + kk * 4);
    }

    // ---- Step 1: each wave computes 16 M-tiles x 2 N-tiles of C = A@B (K=16)
    const size_t o_stride = (size_t)I_ * DOUT * DIN;   // 524288 floats between o slices
    for (int mi = 0; mi < 16; ++mi) {
        const int mtile = wave * 16 + mi;
        const int m = mtile * 16 + l16;      // global row = o*32 + d
        const int o = m >> 5;
        const int d = m & 31;
        const float* wp = W + (size_t)o * o_stride + (size_t)i * (DOUT * DIN)
                            + d * DIN + halfsel * 2;
        // Prefetch next M-tile's W row (gfx1250 global_prefetch_b8)
        if (mi + 1 < 16)
            __builtin_prefetch(wp + 16 * DIN, 0, 1);

        // A fragments: 16x4 f32 per k-step; lane<16 K={c,c+1}, lane>=16 K={c+2,c+3}
        v2f af[4];
#pragma unroll
        for (int kk = 0; kk < 4; ++kk)
            af[kk] = *(const v2f*)(wp + kk * 4);

#pragma unroll
        for (int nt = 0; nt < 2; ++nt) {
            v8f acc = {};
#pragma unroll
            for (int kk = 0; kk < 4; ++kk) {
                acc = __builtin_amdgcn_wmma_f32_16x16x4_f32(
                        /*neg_a=*/false, af[kk],
                        /*neg_b=*/false, bf[nt][kk],
                        /*c_mod=*/(short)0, acc,
                        /*reuse_a=*/false, /*reuse_b=*/false);
            }
            // C/D layout: VGPR j -> row (j + 8*halfsel), col = l16
            const int row0 = mtile * 16 + halfsel * 8;
            const int col  = nt * 16 + l16;
            float* dst = &uhat[row0 * LDS_STRIDE + col];
#pragma unroll
            for (int j = 0; j < 8; ++j)
                dst[j * LDS_STRIDE] = acc[j];
        }
    }

    __syncthreads();

    // ---- Step 2: per (b,d): softmax over o of z = u*c, accumulate s[b,o,d] += p*u
    for (int p = tid; p < B_ * DOUT; p += 256) {
        const int bb = p >> 5;      // batch
        const int dd = p & 31;      // out dim
        if (iter == 0) {
            // c == 0 => uniform routing probs = 1/O
            const float inv_o = 1.0f / (float)O_;
#pragma unroll 4
            for (int o = 0; o < O_; ++o) {
                const float u = uhat[(o * DOUT + dd) * LDS_STRIDE + bb];
                unsafeAtomicAdd(&s[bb * (O_ * DOUT) + o * DOUT + dd], u * inv_o);
            }
        } else {
            const float* cp = c + bb * (O_ * DOUT) + dd;
            float mx = -INFINITY;
#pragma unroll 4
            for (int o = 0; o < O_; ++o) {
                const float u = uhat[(o * DOUT + dd) * LDS_STRIDE + bb];
                mx = fmaxf(mx, u * cp[o * DOUT]);
            }
            float sum = 0.0f;
#pragma unroll 4
            for (int o = 0; o < O_; ++o) {
                const float u = uhat[(o * DOUT + dd) * LDS_STRIDE + bb];
                sum += __expf(u * cp[o * DOUT] - mx);
            }
            const float inv = 1.0f / sum;
#pragma unroll 4
            for (int o = 0; o < O_; ++o) {
                const float u = uhat[(o * DOUT + dd) * LDS_STRIDE + bb];
                const float prob = __expf(u * cp[o * DOUT] - mx) * inv;
                unsafeAtomicAdd(&s[bb * (O_ * DOUT) + o * DOUT + dd], prob * u);
            }
        }
    }
}

// One thread per (b,o): squash s -> v; c += v (iters 0,1) or write v to out (iter 2)
__global__ void caps_squash_kernel(const float* __restrict__ s,
                                   float* __restrict__ c,
                                   float* __restrict__ out,
                                   int iter)
{
    const int idx = blockIdx.x * blockDim.x + threadIdx.x;   // b*O + o
    if (idx >= B_ * O_) return;
    const float* sp = s + (size_t)idx * DOUT;
    float n2 = 0.0f;
#pragma unroll
    for (int d = 0; d < DOUT; ++d) {
        const float v = sp[d];
        n2 += v * v;
    }
    const float scale = n2 / ((1.0f + n2) * sqrtf(n2 + EPS_));
#pragma unroll
    for (int d = 0; d < DOUT; ++d) {
        const float v = sp[d] * scale;
        if (iter < 2) c[(size_t)idx * DOUT + d] += v;
        else          out[(size_t)idx * DOUT + d] = v;
    }
}

extern "C" void kernel_launch(void* const* d_in, const int* in_sizes, int n_in,
                              void* d_out, int out_size, void* d_ws, size_t ws_size,
                              hipStream_t stream) {
    (void)in_sizes; (void)n_in; (void)out_size; (void)ws_size;
    const float* x = (const float*)d_in[0];   // [32, 1024, 16]
    const float* W = (const float*)d_in[1];   // [64, 1024, 32, 16]
    float* out = (float*)d_out;               // [32, 64, 32]

    const int SN = B_ * O_ * DOUT;            // 65536
    float* s = (float*)d_ws;                  // [B,O,DOUT] accumulator
    float* c = s + SN;                        // [B,O,DOUT] cumulative squash outputs

    // Allow >64KB dynamic LDS (CDNA5 WGP has 320KB). Host-side, capture-safe.
    hipFuncSetAttribute((const void*)caps_route_pass_kernel,
                        hipFuncAttributeMaxDynamicSharedMemorySize, LDS_BYTES);

    caps_zero_kernel<<<(SN + 255) / 256, 256, 0, stream>>>(c, SN);
    for (int t = 0; t < 3; ++t) {
        caps_zero_kernel<<<(SN + 255) / 256, 256, 0, stream>>>(s, SN);
        caps_route_pass_kernel<<<I_, 256, LDS_BYTES, stream>>>(x, W, c, s, t);
        caps_squash_kernel<<<(B_ * O_ + 255) / 256, 256, 0, stream>>>(s, c, out, t);
    }
}